// MockCoreAttention_55731495633080
// MI455X (gfx1250) — compile-verified
//
#include <hip/hip_runtime.h>
#include <math.h>

// ---------------------------------------------------------------------------
// Flash-attention forward for (s=2048, b=2, n=16, d=128), fp32 in / fp32 out.
// Matrix math on V_WMMA_F32_16X16X4_F32 (full fp32 fidelity to reference).
// K/V tiles double-buffered in LDS via GLOBAL_LOAD_ASYNC_TO_LDS_B128
// (ASYNCcnt-tracked), overlapping the next tile's DMA with current compute.
// ---------------------------------------------------------------------------

typedef __attribute__((ext_vector_type(2))) float v2f;
typedef __attribute__((ext_vector_type(4))) float v4f;
typedef __attribute__((ext_vector_type(8))) float v8f;
typedef __attribute__((ext_vector_type(4))) int   v4i;

#define AS1 __attribute__((address_space(1)))
#define AS3 __attribute__((address_space(3)))

#if __has_builtin(__builtin_amdgcn_global_load_async_to_lds_b128) && \
    __has_builtin(__builtin_amdgcn_s_wait_asynccnt)
#define HAVE_ASYNC_LDS 1
#else
#define HAVE_ASYNC_LDS 0
#endif

#define S_LEN     2048
#define BATCH     2
#define HEADS     16
#define DHEAD     128
#define ROWSTRIDE (BATCH * HEADS * DHEAD)   // 4096 floats between seq positions
#define QTILE     64                         // queries per workgroup (4 waves x 16)
#define KTILE     32                         // keys per main-loop iteration
#define NITER     (S_LEN / KTILE)            // 64
#define LDSK_STR  132                        // padded fp32 row stride for K/V tiles
#define LDSP_STR  36                         // padded fp32 row stride for P scratch

// softmax scale folded into Q, converted to log2 domain for exp2f softmax
#define QSCALE (0.08838834764831845f * 1.44269504088896340736f)

// Stage 32 contiguous floats of one K/V row into LDS.
__device__ __forceinline__ void stage_row32(const float* __restrict__ g,
                                            float* __restrict__ l) {
#if HAVE_ASYNC_LDS
#pragma unroll
    for (int t = 0; t < 32; t += 4)
        __builtin_amdgcn_global_load_async_to_lds_b128(
            (AS1 v4i*)(g + t), (AS3 v4i*)(l + t), 0, 0);
#else
#pragma unroll
    for (int t = 0; t < 32; t += 4)
        *(v4f*)(l + t) = *(const v4f*)(g + t);
#endif
}

__device__ __forceinline__ void stage_wait() {
#if HAVE_ASYNC_LDS
    __builtin_amdgcn_s_wait_asynccnt(0);
#endif
}

__launch_bounds__(128)
__global__ void fa_fwd_f32_wmma(const float* __restrict__ q,
                                const float* __restrict__ k,
                                const float* __restrict__ v,
                                float* __restrict__ out)
{
    __shared__ float ldsK[2][KTILE * LDSK_STR];         // 2 x 16896 B
    __shared__ float ldsV[2][KTILE * LDSK_STR];         // 2 x 16896 B
    __shared__ float ldsP[4 * 16 * LDSP_STR];           //     9216 B

    const int tid  = threadIdx.x;
    const int wave = tid >> 5;
    const int lane = tid & 31;
    const int col  = lane & 15;   // M (A/C row) or N (B/C col) within a 16-tile
    const int hi   = lane >> 4;   // selects which K-pair of the 4-wide K slab

    const int qt   = blockIdx.x;          // 0..31 query tile
    const int bh   = blockIdx.y;          // 0..31 fused (b*HEADS + h)
    const int base = bh * DHEAD;          // flat offset of (b,h) within a seq row

    // per-thread staging assignment: 4 threads per K/V row, 32 floats each
    const int srow = tid >> 2;
    const int scol = (tid & 3) * 32;
    const float* kgbase = k + (size_t)srow * ROWSTRIDE + base + scol;
    const float* vgbase = v + (size_t)srow * ROWSTRIDE + base + scol;
    const int ldst = srow * LDSK_STR + scol;

    // -----------------------------------------------------------------
    // Load this wave's 16 Q rows into A-fragments (32 slabs of K=4),
    // pre-scaled into log2 domain. A layout (32b, 16x4):
    //   lane = hi*16 + m ; vgpr0 = K=2*hi, vgpr1 = K=2*hi+1
    // -----------------------------------------------------------------
    const float* qrow = q + (size_t)(qt * QTILE + wave * 16 + col) * ROWSTRIDE + base;
    v2f afrag[32];
#pragma unroll
    for (int kk = 0; kk < 32; ++kk) {
        const float* p = qrow + kk * 4 + hi * 2;
        v2f a;
        a.x = p[0] * QSCALE;
        a.y = p[1] * QSCALE;
        afrag[kk] = a;
    }

    // O accumulator: 8 d-tiles of 16x16 in C layout (row = r + 8*hi, col = dt*16+col)
    v8f o[8];
#pragma unroll
    for (int i = 0; i < 8; ++i) o[i] = (v8f){};
    float mrow[8], lrow[8];
#pragma unroll
    for (int r = 0; r < 8; ++r) { mrow[r] = -1.0e30f; lrow[r] = 0.0f; }

    // ---- prologue: stage tile 0 into buffer 0 ----
    stage_row32(kgbase, &ldsK[0][ldst]);
    stage_row32(vgbase, &ldsV[0][ldst]);
    stage_wait();
    __syncthreads();

    for (int j = 0; j < NITER; ++j) {
        const int cur = j & 1;

        // ---- overlap: kick off next tile's global->LDS DMA before compute ----
        if (j + 1 < NITER) {
            const size_t adv = (size_t)(j + 1) * KTILE * ROWSTRIDE;
            stage_row32(kgbase + adv, &ldsK[cur ^ 1][ldst]);
            stage_row32(vgbase + adv, &ldsV[cur ^ 1][ldst]);
        }

        const float* ldsKc = ldsK[cur];
        const float* ldsVc = ldsV[cur];

        // ---- S = (Q*scale) . K^T : two 16x16 tiles, K-dim 128 in slabs of 4 ----
        v8f st[2];
#pragma unroll
        for (int nt = 0; nt < 2; ++nt) {
            v8f acc = (v8f){};
            const float* kb = ldsKc + (nt * 16 + col) * LDSK_STR + hi * 2;
#pragma unroll
            for (int kk = 0; kk < 32; ++kk) {
                v2f b = *(const v2f*)(kb + kk * 4);   // ds_load_b64
                acc = __builtin_amdgcn_wmma_f32_16x16x4_f32(
                        false, afrag[kk], false, b, (short)0, acc, false, false);
            }
            st[nt] = acc;
        }

        // ---- online softmax (log2 domain), stats per C-layout row r+8*hi ----
        float alpha[8];
#pragma unroll
        for (int r = 0; r < 8; ++r) {
            float mx = fmaxf(st[0][r], st[1][r]);
#pragma unroll
            for (int off = 8; off >= 1; off >>= 1)
                mx = fmaxf(mx, __shfl_xor(mx, off, 32));
            const float mnew = fmaxf(mrow[r], mx);
            alpha[r] = exp2f(mrow[r] - mnew);
            mrow[r]  = mnew;
            const float p0 = exp2f(st[0][r] - mnew);
            const float p1 = exp2f(st[1][r] - mnew);
            st[0][r] = p0;
            st[1][r] = p1;
            float rs = p0 + p1;
#pragma unroll
            for (int off = 8; off >= 1; off >>= 1)
                rs += __shfl_xor(rs, off, 32);
            lrow[r] = lrow[r] * alpha[r] + rs;
        }

        // ---- rescale O, spill P (C layout) to per-wave LDS scratch ----
        float* pw = ldsP + wave * 16 * LDSP_STR;
#pragma unroll
        for (int r = 0; r < 8; ++r) {
            const int m = r + 8 * hi;
            pw[m * LDSP_STR + col]      = st[0][r];
            pw[m * LDSP_STR + 16 + col] = st[1][r];
#pragma unroll
            for (int dt = 0; dt < 8; ++dt) o[dt][r] *= alpha[r];
        }

        // ---- reload P as A-fragments (same-wave DS ordering) ----
        const float* pr = pw + col * LDSP_STR + hi * 2;
        v2f pf[8];
#pragma unroll
        for (int kk = 0; kk < 8; ++kk) pf[kk] = *(const v2f*)(pr + kk * 4);

        // ---- O += P . V : 8 d-tiles, K-dim 32 in slabs of 4 ----
#pragma unroll
        for (int dt = 0; dt < 8; ++dt) {
            v8f acc = o[dt];
#pragma unroll
            for (int kk = 0; kk < 8; ++kk) {
                const int kr = kk * 4 + hi * 2;
                v2f bf;
                bf.x = ldsVc[kr * LDSK_STR + dt * 16 + col];
                bf.y = ldsVc[(kr + 1) * LDSK_STR + dt * 16 + col];
                acc = __builtin_amdgcn_wmma_f32_16x16x4_f32(
                        false, pf[kk], false, bf, (short)0, acc, false, false);
            }
            o[dt] = acc;
        }

        // ---- close iteration: next-tile DMA done + all waves done reading ----
        stage_wait();
        __syncthreads();
    }

    // ---- normalize and store: out flat layout == s*4096 + bh*128 + d ----
#pragma unroll
    for (int r = 0; r < 8; ++r) {
        const int m = r + 8 * hi;
        const float inv = 1.0f / lrow[r];
        float* orow = out + (size_t)(qt * QTILE + wave * 16 + m) * ROWSTRIDE + base;
#pragma unroll
        for (int dt = 0; dt < 8; ++dt)
            orow[dt * 16 + col] = o[dt][r] * inv;
    }
}

extern "C" void kernel_launch(void* const* d_in, const int* in_sizes, int n_in,
                              void* d_out, int out_size, void* d_ws, size_t ws_size,
                              hipStream_t stream) {
    const float* q = (const float*)d_in[0];
    const float* k = (const float*)d_in[1];
    const float* v = (const float*)d_in[2];
    float* out = (float*)d_out;

    dim3 grid(S_LEN / QTILE, BATCH * HEADS);   // (32, 32)
    dim3 block(128);                           // 4 waves of 32
    fa_fwd_f32_wmma<<<grid, block, 0, stream>>>(q, k, v, out);
}